// Voxelization_33586644254828
// MI455X (gfx1250) — compile-verified
//
#include <hip/hip_runtime.h>
#include <stdint.h>

// Problem constants (fixed by setup_inputs: b=16, c=64, n=65536, r=32)
#define B   16
#define C   64
#define N   65536
#define R   32
#define R3  32768                 // 32*32*32
#define VOX_OUT_ELEMS (B * C * R3)   // 33,554,432 floats
#define CNT_ELEMS     (B * R3)       // 524,288 floats (workspace)

typedef float v4f __attribute__((ext_vector_type(4)));
typedef int   v4i __attribute__((ext_vector_type(4)));

__device__ __forceinline__ float nclamp(float v) {
    return fminf(fmaxf((v + 1.0f) * 16.0f, 0.0f), 31.0f);   // clip((c+1)/2*32, 0, 31)
}

// ---------------------------------------------------------------------------
// K0: zero the per-batch voxel count grid in workspace (128-bit stores)
// ---------------------------------------------------------------------------
__global__ __launch_bounds__(256) void vox_zero_cnt(float* __restrict__ cnt) {
    int t = blockIdx.x * 256 + threadIdx.x;          // < CNT_ELEMS/4
    v4f z = {0.0f, 0.0f, 0.0f, 0.0f};
    if (t < CNT_ELEMS / 4) *((v4f*)cnt + t) = z;
}

// ---------------------------------------------------------------------------
// K1: 4 points per thread -> norm_coords (d_out), voxel indices (ws),
//     count global atomics. Streams are NT (read/written exactly once).
// ---------------------------------------------------------------------------
__global__ __launch_bounds__(256) void vox_index_count(
    const float* __restrict__ coords,   // (B, 3, N)
    float*       __restrict__ norm_out, // (B, 3, N) region of d_out
    int*         __restrict__ idx,      // (B, N) in ws
    float*       __restrict__ cnt)      // (B, R3) in ws
{
    int t = blockIdx.x * 256 + threadIdx.x;          // < B*N/4 = 262144
    if (t >= (B * N) / 4) return;
    int b  = t >> 14;                                // N/4 == 16384 per batch
    int n4 = (t & 16383) << 2;

    const float* cb = coords + (size_t)b * 3 * N + n4;
    v4f x = __builtin_nontemporal_load((const v4f*)(cb));
    v4f y = __builtin_nontemporal_load((const v4f*)(cb + N));
    v4f z = __builtin_nontemporal_load((const v4f*)(cb + 2 * N));

    v4f nx, ny, nz;
    nx.x = nclamp(x.x); nx.y = nclamp(x.y); nx.z = nclamp(x.z); nx.w = nclamp(x.w);
    ny.x = nclamp(y.x); ny.y = nclamp(y.y); ny.z = nclamp(y.z); ny.w = nclamp(y.w);
    nz.x = nclamp(z.x); nz.y = nclamp(z.y); nz.z = nclamp(z.z); nz.w = nclamp(z.w);

    float* no = norm_out + (size_t)b * 3 * N + n4;
    __builtin_nontemporal_store(nx, (v4f*)(no));
    __builtin_nontemporal_store(ny, (v4f*)(no + N));
    __builtin_nontemporal_store(nz, (v4f*)(no + 2 * N));

    // jnp.round = round-half-to-even = v_cvt rne
    v4i id;
    id.x = __float2int_rn(nx.x) * (R * R) + __float2int_rn(ny.x) * R + __float2int_rn(nz.x);
    id.y = __float2int_rn(nx.y) * (R * R) + __float2int_rn(ny.y) * R + __float2int_rn(nz.y);
    id.z = __float2int_rn(nx.z) * (R * R) + __float2int_rn(ny.z) * R + __float2int_rn(nz.z);
    id.w = __float2int_rn(nx.w) * (R * R) + __float2int_rn(ny.w) * R + __float2int_rn(nz.w);

    *(v4i*)(idx + (size_t)b * N + n4) = id;          // re-read 64x later: keep RT

    float* cg = cnt + (size_t)b * R3;
    atomicAdd(cg + id.x, 1.0f);                      // global_atomic_add_f32
    atomicAdd(cg + id.y, 1.0f);
    atomicAdd(cg + id.z, 1.0f);
    atomicAdd(cg + id.w, 1.0f);
}

// ---------------------------------------------------------------------------
// K3: one workgroup per (batch, channel). Full 32^3 channel accumulator in
// 128 KB LDS (CDNA5-only: 320 KB/WGP -> 2 WGs co-resident). Per tile each
// thread handles 4 points: indices double-buffered via async-to-LDS b128
// (ASYNCcnt), features via NT b128 loads, accumulation via ds_add_f32, and
// the flush fuses the count division with NT b128 stores. Zero global
// atomics on the 256 MB feature path.
// ---------------------------------------------------------------------------
#define TPB  512
#define PPT  4                    // points per thread per tile
#define TILE (TPB * PPT)          // 2048 points
#define NT_TILES (N / TILE)       // 32 tiles

__global__ __launch_bounds__(TPB) void vox_scatter_channel(
    const float* __restrict__ feat,   // (B, C, N)
    const int*   __restrict__ idx,    // (B, N)
    const float* __restrict__ cnt,    // (B, R3)
    float*       __restrict__ out)    // (B, C, R3)
{
    __shared__ float acc[R3];          // 128 KB accumulator
    __shared__ int   ibuf[2][TILE];    // 16 KB async index staging (double buf)

    const int tid = threadIdx.x;
    const int b   = blockIdx.x >> 6;   // C == 64
    const int c   = blockIdx.x & 63;

    const float* f  = feat + ((size_t)(b * C + c)) * N;
    const int*   ix = idx  + (size_t)b * N;

    // zero the accumulator
    for (int i = tid; i < R3; i += TPB) acc[i] = 0.0f;

    // stage tile 0 asynchronously (each lane loads its own 4 slots, b128)
    {
        unsigned lo = (unsigned)(uintptr_t)&ibuf[0][tid * PPT];
        const int* g = ix + tid * PPT;
        asm volatile("global_load_async_to_lds_b128 %0, %1, off"
                     :: "v"(lo), "v"(g) : "memory");
    }
    __syncthreads();

    for (int t = 0; t < NT_TILES; ++t) {
        const int buf = t & 1;
        if (t + 1 < NT_TILES) {
            unsigned lo = (unsigned)(uintptr_t)&ibuf[buf ^ 1][tid * PPT];
            const int* g = ix + (t + 1) * TILE + tid * PPT;
            asm volatile("global_load_async_to_lds_b128 %0, %1, off"
                         :: "v"(lo), "v"(g) : "memory");
            __builtin_prefetch(f + (t + 1) * TILE + tid * PPT, 0, 3);
            asm volatile("s_wait_asynccnt 0x1" ::: "memory");   // tile t arrived
        } else {
            asm volatile("s_wait_asynccnt 0x0" ::: "memory");
        }
        v4f fv = __builtin_nontemporal_load(
                     (const v4f*)(f + t * TILE + tid * PPT));   // read-once stream
        v4i vi = *(const v4i*)&ibuf[buf][tid * PPT];            // ds_load_b128
        atomicAdd(&acc[vi.x], fv.x);                            // ds_add_f32
        atomicAdd(&acc[vi.y], fv.y);
        atomicAdd(&acc[vi.z], fv.z);
        atomicAdd(&acc[vi.w], fv.w);
    }
    __syncthreads();

    // flush: out = acc / (cnt == 0 ? 1e-5 : cnt), NT b128 stores
    float*       op = out + ((size_t)(b * C + c)) * R3;
    const float* cb = cnt + (size_t)b * R3;
    for (int v = tid * 4; v < R3; v += TPB * 4) {
        v4f cv = *(const v4f*)(cb + v);        // reused 64x per batch: RT
        v4f av = *(const v4f*)(&acc[v]);       // ds_load_b128
        v4f r;
        r.x = av.x / ((cv.x == 0.0f) ? 1e-5f : cv.x);
        r.y = av.y / ((cv.y == 0.0f) ? 1e-5f : cv.y);
        r.z = av.z / ((cv.z == 0.0f) ? 1e-5f : cv.z);
        r.w = av.w / ((cv.w == 0.0f) ? 1e-5f : cv.w);
        __builtin_nontemporal_store(r, (v4f*)(op + v));
    }
}

// ---------------------------------------------------------------------------
extern "C" void kernel_launch(void* const* d_in, const int* in_sizes, int n_in,
                              void* d_out, int out_size, void* d_ws, size_t ws_size,
                              hipStream_t stream) {
    (void)in_sizes; (void)n_in; (void)out_size; (void)ws_size;

    const float* feat   = (const float*)d_in[0];   // (16, 64, 65536) f32
    const float* coords = (const float*)d_in[1];   // (16, 3, 65536) f32
    // d_in[2] = resolution (32), fixed by the problem; hardcoded for LDS sizing.

    float* out      = (float*)d_out;               // vox_feature region
    float* norm_out = out + VOX_OUT_ELEMS;         // norm_coords region

    float* cnt = (float*)d_ws;                     // (B, R3) counts
    int*   idx = (int*)d_ws + CNT_ELEMS;           // (B, N) voxel indices

    vox_zero_cnt<<<CNT_ELEMS / 4 / 256, 256, 0, stream>>>(cnt);
    vox_index_count<<<(B * N) / 4 / 256, 256, 0, stream>>>(coords, norm_out, idx, cnt);
    vox_scatter_channel<<<B * C, TPB, 0, stream>>>(feat, idx, cnt, out);
}